// OffsetLoRAModel_31602369364691
// MI455X (gfx1250) — compile-verified
//
#include <hip/hip_runtime.h>
#include <hip/hip_bf16.h>

// out = x @ ((B@A - B0@A0) * s)  ==  (x @ [B | B0]) @ [s*A ; -s*A0]
// Rank-32 factored path: HBM-bound (~256 MiB -> ~11us @ 23.3 TB/s), so keep
// everything in f32 and use V_WMMA_F32_16X16X4_F32 for the matrix math.
//
// v2 changes vs v1:
//  * Cp/Dp are packed pairwise (K-pairs adjacent) so every WMMA B operand is
//    a single aligned 8B load (b64), contiguous 128B across a half-wave.
//  * Kernel 2 splits K 8-ways per block (4096 waves stream x) and reduces the
//    eight partial 16x32 tiles through LDS in a fixed, deterministic order.
//  * Kernel 3 stages the 32KB Dp slab in LDS once per block (8 waves share),
//    WMMA B operands come from ds_load_b64; output stream stays the limiter.

typedef __attribute__((ext_vector_type(2))) float v2f;
typedef __attribute__((ext_vector_type(8))) float v8f;

#define LORA_R   16
#define LORA_SC  1.0f   // lora_alpha / r = 16/16

// ---------------------------------------------------------------------------
// Kernel 1: pack
//   Cp: pairwise [d/2][32][2]:  Cp[(k>>1)*64 + j*2 + (k&1)] = [B|B0][k][j]
//   Dp: pairwise [16][d][2]:    Dp[(j>>1)*2d + k*2 + (j&1)] = [s*A;-s*A0][j][k]
// ---------------------------------------------------------------------------
__global__ __launch_bounds__(256)
void lora_pack(const float* __restrict__ A,  const float* __restrict__ B,
               const float* __restrict__ A0, const float* __restrict__ B0,
               float* __restrict__ Cp, float* __restrict__ Dp, int d) {
  int idx = blockIdx.x * blockDim.x + threadIdx.x;   // 0 .. d*32-1
  if (idx >= d * 32) return;
  int k = idx >> 5;        // 0..d-1
  int j = idx & 31;        // 0..31
  float cv = (j < LORA_R) ? B[k * LORA_R + j] : B0[k * LORA_R + (j - LORA_R)];
  Cp[(size_t)(k >> 1) * 64 + j * 2 + (k & 1)] = cv;
  float dv = (j < LORA_R) ? ( LORA_SC * A [(size_t)j            * d + k])
                          : (-LORA_SC * A0[(size_t)(j - LORA_R) * d + k]);
  Dp[(size_t)(j >> 1) * 2 * d + k * 2 + (j & 1)] = dv;
}

// ---------------------------------------------------------------------------
// Kernel 2: T[N x 32] = x[N x d] @ C[d x 32]
// One block (8 waves) per 16-token tile; wave w reduces K chunk [w*d/8, ...).
// 4 independent WMMA accumulator chains per wave; partials summed via LDS in
// fixed order (deterministic).
// ---------------------------------------------------------------------------
__global__ __launch_bounds__(256)
void lora_xc(const float* __restrict__ x, const float* __restrict__ Cp,
             float* __restrict__ T, int d) {
  __shared__ float red[8 * 512];   // 8 partial 16x32 tiles, 16 KB

  const int w    = threadIdx.x >> 5;     // wave in block: K chunk id
  const int lane = threadIdx.x & 31;
  const int half = lane >> 4;            // 0: K pair {0,1}, 1: K pair {2,3}
  const int ln   = lane & 15;

  const int m0     = blockIdx.x * 16;    // token tile base
  const int dchunk = d >> 3;             // 512 K values per wave
  const int kbase  = w * dchunk;

  const float* xrow   = x  + (size_t)(m0 + ln) * d + kbase + 2 * half;
  const float* cpBase = Cp + (size_t)(kbase >> 1) * 64;

  v8f acc0a = {}; v8f acc0b = {};        // K stream 0, cols 0-15 / 16-31
  v8f acc1a = {}; v8f acc1b = {};        // K stream 1

  for (int k0 = 0; k0 < dchunk; k0 += 8) {
    // A operand (x rows): lane ln holds row m0+ln, K = kbase+k0+2*half+{0,1}
    v2f a0 = *(const v2f*)(xrow + k0);
    v2f a1 = *(const v2f*)(xrow + k0 + 4);

    const int p0 = (k0 >> 1) + half;     // K-pair index, stream 0
    const int p1 = (k0 >> 1) + 2 + half; // K-pair index, stream 1
    v2f b00 = *(const v2f*)(cpBase + (size_t)p0 * 64 + ln * 2);
    v2f b01 = *(const v2f*)(cpBase + (size_t)p0 * 64 + (16 + ln) * 2);
    v2f b10 = *(const v2f*)(cpBase + (size_t)p1 * 64 + ln * 2);
    v2f b11 = *(const v2f*)(cpBase + (size_t)p1 * 64 + (16 + ln) * 2);

    acc0a = __builtin_amdgcn_wmma_f32_16x16x4_f32(false, a0, false, b00, (short)0, acc0a, false, false);
    acc0b = __builtin_amdgcn_wmma_f32_16x16x4_f32(false, a0, false, b01, (short)0, acc0b, false, false);
    acc1a = __builtin_amdgcn_wmma_f32_16x16x4_f32(false, a1, false, b10, (short)0, acc1a, false, false);
    acc1b = __builtin_amdgcn_wmma_f32_16x16x4_f32(false, a1, false, b11, (short)0, acc1b, false, false);
  }

  v8f ta = acc0a + acc1a;
  v8f tb = acc0b + acc1b;

  // C/D layout: VGPR v -> row M = v + 8*half, col = ln
#pragma unroll
  for (int v = 0; v < 8; ++v) {
    red[w * 512 + (v + 8 * half) * 32 + ln]      = ta[v];
    red[w * 512 + (v + 8 * half) * 32 + 16 + ln] = tb[v];
  }
  __syncthreads();

  // Deterministic tree-free reduction: 512 elements, fixed w order.
  for (int e = threadIdx.x; e < 512; e += 256) {
    float s = 0.0f;
#pragma unroll
    for (int w2 = 0; w2 < 8; ++w2) s += red[w2 * 512 + e];
    T[(size_t)m0 * 32 + e] = s;          // T row-major [N][32]
  }
}

// ---------------------------------------------------------------------------
// Kernel 3: out[N x d] = T[N x 32] @ D[32 x d]
// Block = 8 waves sharing one 256-column group; Dp slab (32KB) staged in LDS
// once per block; each wave owns a 16-token tile and emits 16 WMMA tiles.
// ---------------------------------------------------------------------------
__global__ __launch_bounds__(256)
void lora_td(const float* __restrict__ T, const float* __restrict__ Dp,
             float* __restrict__ out, int d, int colGroups) {
  __shared__ float Ds[16 * 512];   // [pair p][colLocal*2 + v], 32 KB

  const int cg     = blockIdx.x % colGroups;        // 256-col group
  const int mtBase = (blockIdx.x / colGroups) * 8;  // 8 token tiles per block

  // Cooperative staging of Dp slab: 16 pairs x 512 floats, b128 coalesced.
  for (int i = threadIdx.x; i < 16 * 128; i += 256) {
    const int p = i >> 7;          // pair 0..15
    const int q = i & 127;         // float4 index within 512-float segment
    ((float4*)Ds)[p * 128 + q] =
        *(const float4*)(Dp + (size_t)p * 2 * d + (size_t)cg * 512 + q * 4);
  }
  __syncthreads();

  const int w    = threadIdx.x >> 5;
  const int lane = threadIdx.x & 31;
  const int half = lane >> 4;
  const int ln   = lane & 15;
  const int m0   = (mtBase + w) * 16;

  // Preload the whole A operand (T tile, 16x32) for all 8 K-steps.
  const float* trow = T + (size_t)(m0 + ln) * 32 + 2 * half;
  v2f a[8];
#pragma unroll
  for (int kk = 0; kk < 8; ++kk) a[kk] = *(const v2f*)(trow + kk * 4);

  for (int ct = 0; ct < 16; ++ct) {
    const int n0 = cg * 256 + ct * 16;
    v8f c = {};
#pragma unroll
    for (int kk = 0; kk < 8; ++kk) {
      const int p = kk * 2 + half;           // K-pair index (kr = kk*4+2*half)
      v2f b = *(const v2f*)(Ds + p * 512 + (ct * 16 + ln) * 2);  // ds_load_b64
      c = __builtin_amdgcn_wmma_f32_16x16x4_f32(false, a[kk], false, b, (short)0, c, false, false);
    }
#pragma unroll
    for (int v = 0; v < 8; ++v)
      out[(size_t)(m0 + v + 8 * half) * d + n0 + ln] = c[v];
  }
}

// ---------------------------------------------------------------------------
extern "C" void kernel_launch(void* const* d_in, const int* in_sizes, int n_in,
                              void* d_out, int out_size, void* d_ws, size_t ws_size,
                              hipStream_t stream) {
  const float* x  = (const float*)d_in[0];
  const float* A  = (const float*)d_in[1];
  const float* B  = (const float*)d_in[2];
  const float* A0 = (const float*)d_in[3];
  const float* B0 = (const float*)d_in[4];

  const int d = in_sizes[1] / LORA_R;   // 4096
  const int N = in_sizes[0] / d;        // 8192

  // Workspace layout: Cp (d x 32), Dp (32 x d), T (N x 32) -- 2 MB total.
  float* Cp = (float*)d_ws;
  float* Dp = Cp + (size_t)d * 32;
  float* T  = Dp + (size_t)d * 32;

  // 1) pack Cp and Dp (WMMA-ready pairwise layouts)
  {
    int threads = d * 32;
    lora_pack<<<(threads + 255) / 256, 256, 0, stream>>>(A, B, A0, B0, Cp, Dp, d);
  }
  // 2) T = x @ C   (one block per 16-token tile, 8-way split-K + LDS reduce)
  {
    lora_xc<<<N / 16, 256, 0, stream>>>(x, Cp, T, d);
  }
  // 3) out = T @ D (8 waves per block share a 256-col Dp slab in LDS)
  {
    int colGroups = d / 256;                      // 16
    int blocks = (N / 16 / 8) * colGroups;        // 64 * 16 = 1024
    lora_td<<<blocks, 256, 0, stream>>>(T, Dp, (float*)d_out, d, colGroups);
  }
}